// Discriminator_30709016167120
// MI455X (gfx1250) — compile-verified
//
#include <hip/hip_runtime.h>

typedef float v2f __attribute__((ext_vector_type(2)));
typedef float v8f __attribute__((ext_vector_type(8)));

#define NODE_NUM 128
#define BATCH    32
#define SEQ_LEN  1024
#define HDIM     4
#define CHUNK_T  128                 // timesteps per LDS chunk
#define NSEQ     64                  // sequences per node (BATCH*2)
#define ROWF     (CHUNK_T * 4 + 4)   // floats per sequence row in LDS (+pad -> bank spread)
#define PF       16                  // prefetch depth (loads in flight per buffer)

// Float-index stride between consecutive sequences i -> i+1 of one node:
//   i>>1 selects batch b (stride 2N*T*H floats), i&1 selects s (stride T*H floats)
#define BSTRIDE_F (2 * NODE_NUM * SEQ_LEN * HDIM)   // 1048576
#define SSTRIDE_F (SEQ_LEN * HDIM)                  // 4096

// One block per node. 8 waves project x -> xp into LDS via WMMA f32 16x16x4
// (K=4 == HDIM) with deep load pipelining, then 64 lanes run the ReLU
// recurrence from LDS with h held in VGPRs across chunks.
__global__ __launch_bounds__(256) void rnn_node_kernel(
    const float* __restrict__ x,     // [B, 2N, T, H]
    const float* __restrict__ W_ih,  // [N, H, H]
    const float* __restrict__ W_hh,  // [N, H, H]
    const float* __restrict__ b_ih,  // [N, H]
    const float* __restrict__ b_hh,  // [N, H]
    const float* __restrict__ W_L,   // [1, H]
    float* __restrict__ partials)    // [N]
{
    extern __shared__ float lds[];   // NSEQ * ROWF floats

    const int node = blockIdx.x;
    const int tid  = threadIdx.x;
    const int wave = tid >> 5;
    const int lane = tid & 31;
    const int lm   = lane & 15;      // lane mod 16
    const int hl   = lane >> 4;      // lane half (0/1)

    // Combined bias (added post-WMMA by the storing half-wave).
    const float bias0 = b_ih[node * HDIM + 0] + b_hh[node * HDIM + 0];
    const float bias1 = b_ih[node * HDIM + 1] + b_hh[node * HDIM + 1];
    const float bias2 = b_ih[node * HDIM + 2] + b_hh[node * HDIM + 2];
    const float bias3 = b_ih[node * HDIM + 3] + b_hh[node * HDIM + 3];

    // A matrix: A[m][k] = W_ih[node][m][k] for m<4, else 0.
    // 32-bit A 16x4 layout: lane L holds M=L%16, K=(L<16?0:2)+{0,1}.
    v2f amat;
    {
        const int k0 = hl * 2;
        if (lm < HDIM) {
            amat.x = W_ih[node * 16 + lm * 4 + k0];
            amat.y = W_ih[node * 16 + lm * 4 + k0 + 1];
        } else {
            amat.x = 0.0f;
            amat.y = 0.0f;
        }
    }

    // Recurrence state (threads 0..63 <-> this node's 64 sequences).
    float h0 = 0.f, h1 = 0.f, h2 = 0.f, h3 = 0.f, acc = 0.f;
    float whh[4][4];
    float wl0 = 0.f, wl1 = 0.f, wl2 = 0.f, wl3 = 0.f;
    if (tid < NSEQ) {
        wl0 = W_L[0]; wl1 = W_L[1]; wl2 = W_L[2]; wl3 = W_L[3];
        #pragma unroll
        for (int j = 0; j < 4; ++j)
            #pragma unroll
            for (int i = 0; i < 4; ++i)
                whh[j][i] = W_hh[node * 16 + j * 4 + i];
    }

    for (int c = 0; c < SEQ_LEN / CHUNK_T; ++c) {
        const int t0 = c * CHUNK_T + wave * 16;   // this wave's 16-step slot

        // Per-lane base for sequence i=0 of this node at this time slot.
        // B matrix: B[k][t] = x[t0+t][k]; lane L holds N=L%16, K=(L<16?0:2)+{0,1}
        // -> per-lane float2 at +lm*16B (+8B upper half): 256B fully coalesced.
        const float* xlane = x
            + ((size_t)(2 * node) * SEQ_LEN + (size_t)(t0 + lm)) * HDIM
            + (size_t)(hl * 2);

        v2f b0[PF], b1[PF];

        // Issue PF loads for a batch of sequences [ibase, ibase+PF).
        auto load_batch = [&](v2f* dst, int ibase) {
            #pragma unroll
            for (int k = 0; k < PF; ++k) {
                const int i = ibase + k;
                dst[k] = *(const v2f*)(xlane
                         + (size_t)(i >> 1) * BSTRIDE_F
                         + (size_t)(i & 1) * SSTRIDE_F);
            }
        };
        // WMMA + bias + half-wave b128 store to LDS for a batch.
        auto consume = [&](const v2f* src, int ibase) {
            #pragma unroll
            for (int k = 0; k < PF; ++k) {
                const int i = ibase + k;
                v8f cz = {};
                v8f d = __builtin_amdgcn_wmma_f32_16x16x4_f32(
                    false, amat, false, src[k], (short)0, cz, false, false);
                // D: VGPR r, lanes 0-15 -> row M=r, col N=lane. Rows 0..3 = xp[t][j].
                if (hl == 0) {
                    float4 o;
                    o.x = d[0] + bias0;
                    o.y = d[1] + bias1;
                    o.z = d[2] + bias2;
                    o.w = d[3] + bias3;
                    *(float4*)&lds[(size_t)i * ROWF + (size_t)(wave * 16 + lm) * 4] = o;
                }
            }
        };

        // Double-buffered pipeline: keep 16..32 b64 loads in flight per wave.
        load_batch(b0, 0);
        load_batch(b1, PF);
        consume(b0, 0);
        load_batch(b0, 2 * PF);
        consume(b1, PF);
        load_batch(b1, 3 * PF);
        consume(b0, 2 * PF);
        consume(b1, 3 * PF);

        __syncthreads();

        // --- Scalar ReLU recurrence from LDS, h kept in VGPRs ---
        if (tid < NSEQ) {
            const float4* row = (const float4*)&lds[(size_t)tid * ROWF];
            #pragma unroll 8
            for (int tc = 0; tc < CHUNK_T; ++tc) {
                float4 xp = row[tc];
                float a0 = xp.x + h0 * whh[0][0] + h1 * whh[0][1] + h2 * whh[0][2] + h3 * whh[0][3];
                float a1 = xp.y + h0 * whh[1][0] + h1 * whh[1][1] + h2 * whh[1][2] + h3 * whh[1][3];
                float a2 = xp.z + h0 * whh[2][0] + h1 * whh[2][1] + h2 * whh[2][2] + h3 * whh[2][3];
                float a3 = xp.w + h0 * whh[3][0] + h1 * whh[3][1] + h2 * whh[3][2] + h3 * whh[3][3];
                h0 = fmaxf(a0, 0.0f);
                h1 = fmaxf(a1, 0.0f);
                h2 = fmaxf(a2, 0.0f);
                h3 = fmaxf(a3, 0.0f);
                acc += h0 * wl0 + h1 * wl1 + h2 * wl2 + h3 * wl3;
            }
        }
        __syncthreads();
    }

    // Deterministic in-block reduction of the 64 per-sequence accumulators.
    if (tid < NSEQ) lds[tid] = acc;
    __syncthreads();
    if (tid == 0) {
        float s = 0.0f;
        for (int i = 0; i < NSEQ; ++i) s += lds[i];
        partials[node] = s;
    }
}

// Fixed-order final reduction: deterministic, overwrites poisoned d_out.
__global__ void finalize_kernel(const float* __restrict__ partials,
                                const float* __restrict__ b_L,
                                float* __restrict__ out)
{
    if (blockIdx.x == 0 && threadIdx.x == 0) {
        float s = 0.0f;
        for (int i = 0; i < NODE_NUM; ++i) s += partials[i];
        // out = sum(dot terms) + (T*B*2N) * b_L
        out[0] = s + b_L[0] * (float)(SEQ_LEN * BATCH * 2 * NODE_NUM);
    }
}

extern "C" void kernel_launch(void* const* d_in, const int* in_sizes, int n_in,
                              void* d_out, int out_size, void* d_ws, size_t ws_size,
                              hipStream_t stream)
{
    const float* x    = (const float*)d_in[0];
    const float* W_ih = (const float*)d_in[1];
    const float* W_hh = (const float*)d_in[2];
    const float* b_ih = (const float*)d_in[3];
    const float* b_hh = (const float*)d_in[4];
    const float* W_L  = (const float*)d_in[5];
    const float* b_L  = (const float*)d_in[6];

    float* partials = (float*)d_ws;               // 128 floats = 512 B of scratch
    const size_t shmem = (size_t)NSEQ * ROWF * sizeof(float);  // 132,096 B < 320 KB

    rnn_node_kernel<<<NODE_NUM, 256, shmem, stream>>>(
        x, W_ih, W_hh, b_ih, b_hh, W_L, partials);
    finalize_kernel<<<1, 32, 0, stream>>>(partials, b_L, (float*)d_out);
}